// MultiSE3Transformer_26216480374895
// MI455X (gfx1250) — compile-verified
//
#include <hip/hip_runtime.h>
#include <hip/hip_bf16.h>

#define N_NODES 16384
#define N_EDGES 131072
#define ROW_OUT 960
#define EPSV 1e-5f
#define INV_SQRT3f 0.5773502691896258f
#define INV_SQRT2f 0.7071067811865476f

typedef __attribute__((ext_vector_type(2))) float v2f;
typedef __attribute__((ext_vector_type(8))) float v8f;

// ---------------------------------------------------------------------------
// Kernel 1: zero the 16384x960 accumulator (d_out doubles as the accumulator).
// ---------------------------------------------------------------------------
__global__ __launch_bounds__(256) void zero_f4(float4* __restrict__ p, long n4) {
    long i = (long)blockIdx.x * blockDim.x + threadIdx.x;
    if (i < n4) p[i] = make_float4(0.f, 0.f, 0.f, 0.f);
}

// ---------------------------------------------------------------------------
// Kernel 2: depthwise tensor product + bias + scatter-add. One wave per edge.
// All lane loads are coalesced (channel = lane + 32*i). 960 f32 atomics/edge
// land in the L2-resident 63MB accumulator.
// ---------------------------------------------------------------------------
__global__ __launch_bounds__(256) void tp_scatter(
    const float* __restrict__ node_feat,   // [16384,320]
    const float* __restrict__ edge_sh,     // [E,4]
    const float* __restrict__ edge_weight, // [E,448]
    const float* __restrict__ tp_bias,     // [192]
    const int*   __restrict__ edge_src,
    const int*   __restrict__ edge_dst,
    float*       __restrict__ agg)         // [16384,960]
{
    const int e    = (int)((blockIdx.x * (unsigned)blockDim.x + threadIdx.x) >> 5);
    const int lane = threadIdx.x & 31;

    const int s = edge_src[e];
    const int d = edge_dst[e];
    const float* __restrict__ x   = node_feat + (long)s * 320;
    const float* __restrict__ w   = edge_weight + (long)e * 448;
    float*       __restrict__ out = agg + (long)d * ROW_OUT;

    const float ys  = edge_sh[e * 4 + 0];
    const float yv0 = edge_sh[e * 4 + 1];
    const float yv1 = edge_sh[e * 4 + 2];
    const float yv2 = edge_sh[e * 4 + 3];

    // hint the streaming weight fetch for the next edge row
    __builtin_prefetch(w + 448, 0, 1);

    // ---- scalar-input paths: out0a (128) and out1a (384, at offset 192) ----
#pragma unroll
    for (int i = 0; i < 4; ++i) {
        const int   u  = lane + 32 * i;          // 0..127
        const float xs = x[u];
        // out0a = w0*xs*ys  (+ bias, applied per edge before aggregation)
        atomicAdd(out + u, w[u] * xs * ys + tp_bias[u]);
        // out1a = (w1*xs) outer yv  -> channels 192 + 3u + d
        const float t = w[128 + u] * xs;
        atomicAdd(out + 192 + 3 * u + 0, t * yv0);
        atomicAdd(out + 192 + 3 * u + 1, t * yv1);
        atomicAdd(out + 192 + 3 * u + 2, t * yv2);
    }

    // ---- vector-input paths: out0b (64, off 128), out1b (off 576), out1c (off 768)
#pragma unroll
    for (int i = 0; i < 2; ++i) {
        const int   u   = lane + 32 * i;         // 0..63
        const float xv0 = x[128 + 3 * u + 0];
        const float xv1 = x[128 + 3 * u + 1];
        const float xv2 = x[128 + 3 * u + 2];
        // out0b = w3 * <xv,yv> / sqrt(3)  (+ bias)
        const float dot = xv0 * yv0 + xv1 * yv1 + xv2 * yv2;
        atomicAdd(out + 128 + u, w[320 + u] * dot * INV_SQRT3f + tp_bias[128 + u]);
        // out1b = (w2*ys) * xv
        const float t2 = w[256 + u] * ys;
        atomicAdd(out + 576 + 3 * u + 0, t2 * xv0);
        atomicAdd(out + 576 + 3 * u + 1, t2 * xv1);
        atomicAdd(out + 576 + 3 * u + 2, t2 * xv2);
        // out1c = w4 * (xv x yv) / sqrt(2)
        const float c0 = xv1 * yv2 - xv2 * yv1;
        const float c1 = xv2 * yv0 - xv0 * yv2;
        const float c2 = xv0 * yv1 - xv1 * yv0;
        const float t4 = w[384 + u] * INV_SQRT2f;
        atomicAdd(out + 768 + 3 * u + 0, t4 * c0);
        atomicAdd(out + 768 + 3 * u + 1, t4 * c1);
        atomicAdd(out + 768 + 3 * u + 2, t4 * c2);
    }
}

// ---------------------------------------------------------------------------
// Kernel 3: equivariant LayerNorm, 16 nodes per wave, stats via WMMA.
// ---------------------------------------------------------------------------

// Extract the diagonal of a 16x16 f32 D fragment: node m<8 lives in VGPR m,
// lane m; node m>=8 lives in VGPR m-8, lane m+16. Result: lane m (m<16) holds
// diag[m].
__device__ __forceinline__ float diag16(v8f acc, int lane) {
    float r = 0.f;
#pragma unroll
    for (int j = 0; j < 8; ++j) {
        float v0 = __shfl(acc[j], j, 32);        // node j
        float v1 = __shfl(acc[j], j + 24, 32);   // node j+8 (lane = (j+8)+16)
        r = (lane == j)     ? v0 : r;
        r = (lane == j + 8) ? v1 : r;
    }
    return r;
}

// Reduce channels [4*C0, 4*C1) over 16 node rows: q = per-node sum of squares
// (diag of A*A^T accumulated on the matrix pipe), s = per-node sum (A*ones).
template <int C0, int C1, bool WANT_SUM>
__device__ __forceinline__ void rrange(const float* __restrict__ base, int lane,
                                       float& q, float& s) {
#if __has_builtin(__builtin_amdgcn_wmma_f32_16x16x4_f32)
    const int row  = lane & 15;
    const int koff = (lane >> 4) * 2;            // A 16x4 f32 fragment layout
    v8f aq = {};
    v8f as = {};
    v2f ones; ones.x = 1.0f; ones.y = 1.0f;
    for (int c = C0; c < C1; ++c) {
        const float2 xv = *(const float2*)(base + (long)row * ROW_OUT + 4 * c + koff);
        v2f a; a.x = xv.x; a.y = xv.y;
        aq = __builtin_amdgcn_wmma_f32_16x16x4_f32(false, a, false, a,
                                                   (short)0, aq, false, false);
        if (WANT_SUM)
            as = __builtin_amdgcn_wmma_f32_16x16x4_f32(false, a, false, ones,
                                                       (short)0, as, false, false);
    }
    q = diag16(aq, lane);
    if (WANT_SUM) s = diag16(as, lane);
#else
    // VALU fallback (also used by the host compile pass)
    const int row = lane & 15;
    float qq = 0.f, ss = 0.f;
    for (int k = 4 * C0; k < 4 * C1; ++k) {
        float v = base[(long)row * ROW_OUT + k];
        qq += v * v; ss += v;
    }
    q = qq;
    if (WANT_SUM) s = ss;
#endif
}

__global__ __launch_bounds__(32) void ln_norm(
    float* __restrict__ agg,          // [16384,960], normalized in place
    const float* __restrict__ lnw,    // [448]
    const float* __restrict__ lnb)    // [192]
{
    const int  lane  = threadIdx.x;               // 0..31, full wave active
    const long node0 = (long)blockIdx.x * 16;
    const float* base = agg + node0 * ROW_OUT;

    float dummy;
    float q0, q1, q2, q3, q4, s0, s1;
    rrange<  0,  32, true >(base, lane, q0, s0);  // (128,0) ch [0,128)
    rrange< 32,  48, true >(base, lane, q1, s1);  // (64,0)  ch [128,192)
    rrange< 48, 144, false>(base, lane, q2, dummy); // (128,1) ch [192,576)
    rrange<144, 192, false>(base, lane, q3, dummy); // (64,1)  ch [576,768)
    rrange<192, 240, false>(base, lane, q4, dummy); // (64,1)  ch [768,960)

    for (int m = 0; m < 16; ++m) {
        const float S0 = __shfl(s0, m, 32);
        const float S1 = __shfl(s1, m, 32);
        const float Q0 = __shfl(q0, m, 32);
        const float Q1 = __shfl(q1, m, 32);
        const float Q2 = __shfl(q2, m, 32);
        const float Q3 = __shfl(q3, m, 32);
        const float Q4 = __shfl(q4, m, 32);

        const float mean0 = S0 * (1.f / 128.f);
        const float mean1 = S1 * (1.f / 64.f);
        const float inv0  = rsqrtf(Q0 * (1.f / 128.f) - mean0 * mean0 + EPSV);
        const float inv1  = rsqrtf(Q1 * (1.f / 64.f)  - mean1 * mean1 + EPSV);
        const float inv2  = rsqrtf(Q2 * (1.f / 384.f) + EPSV);
        const float inv3  = rsqrtf(Q3 * (1.f / 192.f) + EPSV);
        const float inv4  = rsqrtf(Q4 * (1.f / 192.f) + EPSV);

        float* __restrict__ rowp = agg + (node0 + m) * ROW_OUT;
#pragma unroll
        for (int j = 0; j < 30; ++j) {
            const int   c = lane + 32 * j;
            const float v = rowp[c];
            float o;
            if (j < 4) {                               // (128,0): center, norm, bias
                o = (v - mean0) * inv0 * lnw[c] + lnb[c];
            } else if (j < 6) {                        // (64,0)
                o = (v - mean1) * inv1 * lnw[c] + lnb[c];
            } else if (j < 18) {                       // (128,1): weight per mul
                const int u = (c - 192) / 3;
                o = v * inv2 * lnw[192 + u];
            } else if (j < 24) {                       // (64,1)
                const int u = (c - 576) / 3;
                o = v * inv3 * lnw[320 + u];
            } else {                                   // (64,1)
                const int u = (c - 768) / 3;
                o = v * inv4 * lnw[384 + u];
            }
            rowp[c] = o;
        }
    }
}

// ---------------------------------------------------------------------------
extern "C" void kernel_launch(void* const* d_in, const int* in_sizes, int n_in,
                              void* d_out, int out_size, void* d_ws, size_t ws_size,
                              hipStream_t stream) {
    const float* node_feat   = (const float*)d_in[0];
    const float* edge_sh     = (const float*)d_in[1];
    const float* edge_weight = (const float*)d_in[2];
    const float* tp_bias     = (const float*)d_in[3];
    const float* ln_weight   = (const float*)d_in[4];
    const float* ln_bias     = (const float*)d_in[5];
    const int*   edge_src    = (const int*)d_in[6];
    const int*   edge_dst    = (const int*)d_in[7];
    float* out = (float*)d_out;

    const long n4 = (long)N_NODES * ROW_OUT / 4;   // 3,932,160 float4
    zero_f4<<<(unsigned)((n4 + 255) / 256), 256, 0, stream>>>((float4*)out, n4);

    tp_scatter<<<N_EDGES / 8, 256, 0, stream>>>(node_feat, edge_sh, edge_weight,
                                                tp_bias, edge_src, edge_dst, out);

    ln_norm<<<N_NODES / 16, 32, 0, stream>>>(out, ln_weight, ln_bias);
}